// FilterNetworkUpsample_6408091206053
// MI455X (gfx1250) — compile-verified
//
#include <hip/hip_runtime.h>
#include <stdint.h>

// Problem constants (from reference: B=8, C=3, H=W=512, 2x upsample, K=5)
#define IN_H   512
#define IN_W   512
#define OUT_H  1024
#define OUT_W  1024

// Block decomposition: 256 threads, each owns a 2x2 output quad.
// Quad grid per block: 32 x 8  -> output tile 64 x 16.
#define QX 32
#define QY 8
#define NTHREADS (QX * QY)          // 256
// Input tile (base 32x8 plus 2-halo each side), 3 channels.
#define TILE_W (QX + 4)             // 36
#define TILE_H (QY + 4)             // 12
#define TILE_PLANE (TILE_H * TILE_W)       // 432
#define TILE_ELEMS (3 * TILE_PLANE)        // 1296
#define STAGE_ROUNDS ((TILE_ELEMS + NTHREADS - 1) / NTHREADS)  // 6

#define LOG2E 1.4426950408889634f

__device__ __forceinline__ float fast_exp2(float x) {
    return __builtin_amdgcn_exp2f(x);               // v_exp_f32
}
__device__ __forceinline__ float fast_rcp(float x) {
    return __builtin_amdgcn_rcpf(x);                // v_rcp_f32
}
__device__ __forceinline__ float fast_sigmoid(float x) {
    // 1 / (1 + e^-x) = 1 / (1 + 2^(-x*log2e))
    return fast_rcp(1.0f + fast_exp2(-LOG2E * x));
}
__device__ __forceinline__ float fast_tanh(float x) {
    // tanh(x) = 2 / (1 + e^{-2x}) - 1
    return 2.0f * fast_rcp(1.0f + fast_exp2(-2.0f * LOG2E * x)) - 1.0f;
}

__global__ __launch_bounds__(NTHREADS)
void fnu_upsample_kernel(const float* __restrict__ input,
                         const float* __restrict__ upar,
                         const float* __restrict__ skip,
                         const float* __restrict__ spar,
                         float* __restrict__ out)
{
    __shared__ float tile[TILE_ELEMS];   // layout [c][y][x], c-major

    const int tid = threadIdx.x;
    const int b   = blockIdx.z;
    const int qx  = tid & (QX - 1);
    const int qy  = tid >> 5;

    // ---------------- Stage input tile into LDS via async copies -----------
    // Global region (per channel): rows [blockIdx.y*QY-2, +TILE_H), cols
    // [blockIdx.x*QX-2, +TILE_W), coordinates clamped (matches jnp.clip).
    const uint64_t inBase = (uint64_t)(input + (size_t)b * 3 * IN_H * IN_W);
    const int bx0 = (int)blockIdx.x * QX - 2;
    const int by0 = (int)blockIdx.y * QY - 2;

    #pragma unroll
    for (int r = 0; r < STAGE_ROUNDS; ++r) {
        int i = tid + r * NTHREADS;
        i = (i < TILE_ELEMS) ? i : (TILE_ELEMS - 1);   // keep EXEC full; dup OK
        int c   = i / TILE_PLANE;
        int rem = i - c * TILE_PLANE;
        int y   = rem / TILE_W;
        int x   = rem - y * TILE_W;
        int gx  = min(max(bx0 + x, 0), IN_W - 1);
        int gy  = min(max(by0 + y, 0), IN_H - 1);
        unsigned goff = (unsigned)(((c * IN_H + gy) * IN_W + gx) * 4);
        // low 32 bits of the flat address of an LDS object == LDS byte offset
        unsigned loff = (unsigned)(uintptr_t)(&tile[i]);
        asm volatile("global_load_async_to_lds_b32 %0, %1, %2"
                     :: "v"(loff), "v"(goff), "s"(inBase)
                     : "memory");
    }
    asm volatile("s_wait_asynccnt 0x0" ::: "memory");
    __syncthreads();

    // ---------------- Per-quad-pixel kernel-shape parameters ----------------
    // Output quad: (oy0..oy0+1, ox0..ox0+1). dx0/dy0 are exactly +-0.25 and
    // both quad columns/rows share bxi/byi, so all 4 pixels share the same
    // 5x5x3 sample footprint already sitting in LDS.
    const int ox0 = (int)blockIdx.x * (2 * QX) + qx * 2;
    const int oy0 = (int)blockIdx.y * (2 * QY) + qy * 2;

    const size_t planeO = (size_t)OUT_H * OUT_W;
    const float* upB  = upar + (size_t)b * 3 * planeO;
    const float* skB  = skip + (size_t)b * 3 * planeO;
    const float* spB  = spar + (size_t)b * planeO;
    float*       outB = out  + (size_t)b * 3 * planeO;

    const float NEG2LOG2E = -2.0f * LOG2E;   // fold exp(-2*dist) -> exp2

    float A2[2][2], B2[2][2], C2[2][2], SP[2][2];
    float acc[2][2][3];
    float wacc[2][2];

    #pragma unroll
    for (int sy = 0; sy < 2; ++sy) {
        #pragma unroll
        for (int sx = 0; sx < 2; ++sx) {
            size_t p = (size_t)(oy0 + sy) * OUT_W + (size_t)(ox0 + sx);
            float kx  = fast_sigmoid(upB[p]) * 4.0f;
            float ky  = fast_sigmoid(upB[planeO + p]) * 4.0f;
            float kxy = fast_tanh(upB[2 * planeO + p]);
            A2[sy][sx] = kx * kx * NEG2LOG2E;               // dx*dx term
            B2[sy][sx] = ky * kx * NEG2LOG2E;               // dy*dy term (ref uses ky*kx)
            C2[sy][sx] = 2.0f * kx * ky * kxy * NEG2LOG2E;  // dx*dy term
            SP[sy][sx] = fast_sigmoid(spB[p]);
            acc[sy][sx][0] = acc[sy][sx][1] = acc[sy][sx][2] = 0.0f;
            wacc[sy][sx] = 0.0f;
        }
    }

    // ---------------- 5x5 taps: all geometry constants fold ----------------
    #pragma unroll
    for (int iy = 0; iy < 5; ++iy) {
        // Tile coords: row = qy + iy (0..11), col = qx + ix (0..35)
        float s[3][5];
        #pragma unroll
        for (int c = 0; c < 3; ++c) {
            #pragma unroll
            for (int ix = 0; ix < 5; ++ix)
                s[c][ix] = tile[c * TILE_PLANE + (qy + iy) * TILE_W + (qx + ix)];
        }
        #pragma unroll
        for (int ix = 0; ix < 5; ++ix) {
            #pragma unroll
            for (int sy = 0; sy < 2; ++sy) {
                #pragma unroll
                for (int sx = 0; sx < 2; ++sx) {
                    const float dx = (sx ? 0.25f : -0.25f) - (float)(ix - 2);
                    const float dy = (sy ? 0.25f : -0.25f) - (float)(iy - 2);
                    // literal dx*dx / dy*dy / dx*dy -> mul + 2 fma + v_exp
                    float w = fast_exp2((dx * dx) * A2[sy][sx]
                                      + (dy * dy) * B2[sy][sx]
                                      + (dx * dy) * C2[sy][sx]);
                    wacc[sy][sx] += w;           // w > 0, abs(w) == w
                    acc[sy][sx][0] += s[0][ix] * w;
                    acc[sy][sx][1] += s[1][ix] * w;
                    acc[sy][sx][2] += s[2][ix] * w;
                }
            }
        }
    }

    // ---------------- Normalize + skip blend, store -------------------------
    #pragma unroll
    for (int sy = 0; sy < 2; ++sy) {
        #pragma unroll
        for (int sx = 0; sx < 2; ++sx) {
            size_t p = (size_t)(oy0 + sy) * OUT_W + (size_t)(ox0 + sx);
            float rw = fast_rcp(wacc[sy][sx] + 1e-6f);
            float sp = SP[sy][sx];
            #pragma unroll
            for (int c = 0; c < 3; ++c) {
                float u = acc[sy][sx][c] * rw;
                outB[c * planeO + p] = u + sp * (skB[c * planeO + p] - u);
            }
        }
    }
}

extern "C" void kernel_launch(void* const* d_in, const int* in_sizes, int n_in,
                              void* d_out, int out_size, void* d_ws, size_t ws_size,
                              hipStream_t stream) {
    const float* input = (const float*)d_in[0];   // (B,3,512,512)
    const float* upar  = (const float*)d_in[1];   // (B,3,1024,1024)
    const float* skip  = (const float*)d_in[2];   // (B,3,1024,1024)
    const float* spar  = (const float*)d_in[3];   // (B,1,1024,1024)
    float* out = (float*)d_out;                   // (B,3,1024,1024)

    const int B = in_sizes[0] / (3 * IN_H * IN_W);
    dim3 grid(OUT_W / (2 * QX),   // 16
              OUT_H / (2 * QY),   // 64
              B);                 // 8
    fnu_upsample_kernel<<<grid, NTHREADS, 0, stream>>>(input, upar, skip, spar, out);
}